// HybridEstimatorQNN_65481071409530
// MI455X (gfx1250) — compile-verified
//
#include <hip/hip_runtime.h>
#include <hip/hip_bf16.h>
#include <math.h>

typedef __attribute__((ext_vector_type(2))) float v2f;
typedef __attribute__((ext_vector_type(8))) float v8f;

// Y[M,N] = act(X[M,K] @ W[N,K]^T + bias[N]); one wave32 per 16x16 tile.
// Uses V_WMMA_F32_16X16X4_F32 (exact fp32, K-step = 4).
template<bool RELU>
__global__ void gemm_wmma_f32(const float* __restrict__ X,
                              const float* __restrict__ W,
                              const float* __restrict__ bias,
                              float* __restrict__ Y,
                              int M, int N, int K)
{
    const int lane  = threadIdx.x & 31;
    const int wave  = (int)((blockIdx.x * blockDim.x + threadIdx.x) >> 5); // wave-uniform
    const int tilesN = N >> 4;
    const int tm = wave / tilesN;
    const int tn = wave - tm * tilesN;
    if (tm * 16 >= M) return;   // wave-uniform guard: EXEC stays all-1s

    const int l15 = lane & 15;
    const int kb  = (lane >> 4) << 1;   // 0 for lanes 0-15, 2 for lanes 16-31

    // A fragment: row = tile_row + (lane&15), k in {kb, kb+1}  -> contiguous float2
    const float* xp = X + (size_t)(tm * 16 + l15) * K + kb;
    // B fragment (W^T): col = tile_col + (lane&15), same k pair -> contiguous float2
    const float* wp = W + (size_t)(tn * 16 + l15) * K + kb;

    v8f acc = {};
#pragma unroll 8
    for (int k = 0; k < K; k += 4) {
        v2f a = *(const v2f*)(xp + k);
        v2f b = *(const v2f*)(wp + k);
        // 8 args: (neg_a, A, neg_b, B, c_mod, C, reuse_a, reuse_b)
        acc = __builtin_amdgcn_wmma_f32_16x16x4_f32(
            false, a, false, b, (short)0, acc, false, false);
    }

    // C/D layout: VGPR r -> row (r + (lane>=16 ? 8 : 0)), col = lane&15
    const int col = tn * 16 + l15;
    const float bn = bias[col];
    const int r0 = tm * 16 + ((lane >> 4) << 3);
#pragma unroll
    for (int r = 0; r < 8; ++r) {
        float v = acc[r] + bn;
        if (RELU) v = v > 0.0f ? v : 0.0f;
        Y[(size_t)(r0 + r) * N + col] = v;
    }
}

// c[i] = H2[i,:] . w3 + b3 ;  out[i] = c + K1*cos(c) + combine_b
// where K1 = mean_w cos(ry[w]) * sum(combine_w)   (analytic collapse of the
// 12-qubit product-state circuit: <Z_w> = cos(ry[w]) * cos(c)).
__global__ void final_head(const float* __restrict__ H2,   // (M,128)
                           const float* __restrict__ w3,   // (128)
                           const float* __restrict__ b3,   // (1)
                           const float* __restrict__ ry,   // (12)
                           const float* __restrict__ cw,   // (4)
                           const float* __restrict__ cb,   // (1)
                           float* __restrict__ out, int M)
{
    int i = blockIdx.x * blockDim.x + threadIdx.x;
    if (i >= M) return;

    const float* h = H2 + (size_t)i * 128;
    float acc = 0.0f;
#pragma unroll
    for (int k = 0; k < 128; ++k)
        acc = fmaf(h[k], w3[k], acc);     // w3 index uniform -> scalar loads
    float c = acc + b3[0];

    float s = 0.0f;
#pragma unroll
    for (int w = 0; w < 12; ++w) s += cosf(ry[w]);
    float k1 = (s * (1.0f / 12.0f)) * (cw[0] + cw[1] + cw[2] + cw[3]);

    out[i] = c + k1 * cosf(c) + cb[0];
}

extern "C" void kernel_launch(void* const* d_in, const int* in_sizes, int n_in,
                              void* d_out, int out_size, void* d_ws, size_t ws_size,
                              hipStream_t stream) {
    const float* x  = (const float*)d_in[0];   // (2048, 64)
    const float* w1 = (const float*)d_in[1];   // (256, 64)
    const float* b1 = (const float*)d_in[2];   // (256)
    const float* w2 = (const float*)d_in[3];   // (128, 256)
    const float* b2 = (const float*)d_in[4];   // (128)
    const float* w3 = (const float*)d_in[5];   // (1, 128)
    const float* b3 = (const float*)d_in[6];   // (1)
    const float* ry = (const float*)d_in[7];   // (12)
    const float* cw = (const float*)d_in[8];   // (1, 4)
    const float* cb = (const float*)d_in[9];   // (1)
    float* out = (float*)d_out;                // (2048,1,1) flat

    const int B = 2048, D = 64, H1n = 256, H2n = 128;

    float* H1 = (float*)d_ws;                     // 2048*256 f32 = 2 MB
    float* H2 = H1 + (size_t)B * H1n;             // 2048*128 f32 = 1 MB

    // Layer 1: (2048x64)@(64x256) + relu. 128*16 tiles -> 2048 waves.
    {
        int waves   = (B / 16) * (H1n / 16);
        int threads = waves * 32;
        gemm_wmma_f32<true><<<threads / 256, 256, 0, stream>>>(x, w1, b1, H1, B, H1n, D);
    }
    // Layer 2: (2048x256)@(256x128) + relu. 128*8 tiles -> 1024 waves.
    {
        int waves   = (B / 16) * (H2n / 16);
        int threads = waves * 32;
        gemm_wmma_f32<true><<<threads / 256, 256, 0, stream>>>(H1, w2, b2, H2, B, H2n, H1n);
    }
    // Layer 3 + analytic quantum epilogue.
    final_head<<<B / 256, 256, 0, stream>>>(H2, w3, b3, ry, cw, cb, out, B);
}